// FastRCNNOutputLayers_79147657331099
// MI455X (gfx1250) — compile-verified
//
#include <hip/hip_runtime.h>
#include <stdint.h>

#define R_ROWS   4000
#define K_CLS    46
#define MAX_CAND 8192
#define TOPK     100
#define NFLAT    (R_ROWS * K_CLS)   // 184000
#define NSORT    262144             // 2^18 >= NFLAT
#define NEGV     (-1e30f)
#define SCALE_CLAMPF 4.135166556742356f
#define IMG_WF   1333.0f
#define IMG_HF   800.0f
#define TJ       512                // j-tile (boxes) held in LDS

typedef float v2f __attribute__((ext_vector_type(2)));
typedef float v8f __attribute__((ext_vector_type(8)));
typedef int   gv4i __attribute__((vector_size(16)));   // matches builtin prototype

#if defined(__has_builtin)
#  if __has_builtin(__builtin_amdgcn_wmma_f32_16x16x4_f32)
#    define HAVE_WMMA4 1
#  endif
#  if __has_builtin(__builtin_amdgcn_global_load_async_to_lds_b128)
#    define HAVE_ASYNC128 1
#  endif
#  if __has_builtin(__builtin_amdgcn_s_wait_asynccnt)
#    define HAVE_WAITASYNC 1
#  endif
#endif

__device__ __forceinline__ bool finitef(float x) {
    return (x == x) && (fabsf(x) <= 3.0e38f);
}
__device__ __forceinline__ unsigned mono32(float f) {
    unsigned b = __float_as_uint(f);
    return (b & 0x80000000u) ? ~b : (b | 0x80000000u);
}
__device__ __forceinline__ float unmono32(unsigned m) {
    unsigned b = (m & 0x80000000u) ? (m & 0x7FFFFFFFu) : ~m;
    return __uint_as_float(b);
}

// 16B global -> LDS, async path on gfx1250 toolchains that expose it.
// Builtin signature (probed via diagnostic): (v4i* AS1, v4i* AS3, Ii, Ii).
__device__ __forceinline__ void copy16_to_lds(const void* g, void* l) {
#ifdef HAVE_ASYNC128
    __builtin_amdgcn_global_load_async_to_lds_b128(
        (__attribute__((address_space(1))) gv4i*)(unsigned long long)g,
        (__attribute__((address_space(3))) gv4i*)(unsigned)(unsigned long long)l,
        0, 0);
#else
    *(float4*)l = *(const float4*)g;
#endif
}
__device__ __forceinline__ void wait_async_copies() {
#ifdef HAVE_ASYNC128
#  ifdef HAVE_WAITASYNC
    __builtin_amdgcn_s_wait_asynccnt(0);
#  else
    asm volatile("s_wait_asynccnt 0" ::: "memory");
#  endif
#endif
}

// ---------------- K1: decode + softmax + sort keys -------------------------
__global__ void k_decode(const float* __restrict__ prop,
                         const float* __restrict__ deltas,
                         const float* __restrict__ logits,
                         float* __restrict__ boxes,
                         unsigned long long* __restrict__ keys) {
    const int r = blockIdx.x;
    const int t = threadIdx.x;
    __shared__ float sl[K_CLS + 1];
    __shared__ float smax, sinv;
    __shared__ int sfin;
    if (t == 0) sfin = 1;
    if (t < K_CLS + 1) sl[t] = logits[(size_t)r * (K_CLS + 1) + t];
    __syncthreads();
    if (t == 0) {
        float mx = sl[0];
        for (int c = 1; c < K_CLS + 1; ++c) mx = fmaxf(mx, sl[c]);
        float sm = 0.f;
        for (int c = 0; c < K_CLS + 1; ++c) sm += expf(sl[c] - mx);
        smax = mx; sinv = 1.0f / sm;
    }
    __syncthreads();
    float score = 0.f;
    if (t < K_CLS) {
        score = expf(sl[t] - smax) * sinv;
        const float p0 = prop[r*4+0], p1 = prop[r*4+1];
        const float p2 = prop[r*4+2], p3 = prop[r*4+3];
        float w = p2 - p0, h = p3 - p1;
        float cx = p0 + 0.5f*w, cy = p1 + 0.5f*h;
        const float* d = deltas + ((size_t)r * K_CLS + t) * 4;
        float dx = d[0] * 0.1f, dy = d[1] * 0.1f;
        float dw = fminf(d[2] * 0.2f, SCALE_CLAMPF);
        float dh = fminf(d[3] * 0.2f, SCALE_CLAMPF);
        float pcx = dx*w + cx, pcy = dy*h + cy;
        float pw = expf(dw)*w, ph = expf(dh)*h;
        bool fin = finitef(pcx) && finitef(pcy) && finitef(pw) && finitef(ph) && finitef(score);
        float x1 = fminf(fmaxf(pcx - 0.5f*pw, 0.f), IMG_WF);
        float y1 = fminf(fmaxf(pcy - 0.5f*ph, 0.f), IMG_HF);
        float x2 = fminf(fmaxf(pcx + 0.5f*pw, 0.f), IMG_WF);
        float y2 = fminf(fmaxf(pcy + 0.5f*ph, 0.f), IMG_HF);
        size_t bo = ((size_t)r * K_CLS + t) * 4;
        boxes[bo+0]=x1; boxes[bo+1]=y1; boxes[bo+2]=x2; boxes[bo+3]=y2;
        if (!fin) atomicAnd(&sfin, 0);
    }
    __syncthreads();
    if (t < K_CLS) {
        bool valid = (sfin != 0) && (score > 0.1f);
        float val = valid ? score : NEGV;
        unsigned idx = (unsigned)(r * K_CLS + t);
        keys[idx] = ((unsigned long long)mono32(val) << 32) | (unsigned)(~idx);
    }
}

__global__ void k_pad(unsigned long long* __restrict__ keys) {
    int i = NFLAT + blockIdx.x * blockDim.x + threadIdx.x;
    if (i < NSORT) keys[i] = 0ull;
}

// ---------------- K2: bitonic sort (descending) ----------------------------
__global__ void k_bitonic(unsigned long long* __restrict__ keys, int j, int k) {
    int i = blockIdx.x * blockDim.x + threadIdx.x;
    int p = i ^ j;
    if (p > i) {
        unsigned long long a = keys[i], b = keys[p];
        bool up = (i & k) == 0;
        if (up ? (a < b) : (a > b)) { keys[i] = b; keys[p] = a; }
    }
}

// ---------------- K3: gather top-8192 candidates ---------------------------
__global__ void k_gather(const unsigned long long* __restrict__ keys,
                         const float* __restrict__ boxes,
                         float* __restrict__ cboxR, float* __restrict__ cboxO,
                         float* __restrict__ cscore, int* __restrict__ ccls,
                         int* __restrict__ cvalid) {
    int t = blockIdx.x * blockDim.x + threadIdx.x;
    if (t >= MAX_CAND) return;
    unsigned long long key = keys[t];
    unsigned idx = ~(unsigned)key;
    if (idx >= NFLAT) idx = 0;                    // defensive (never in top-8192)
    float sc = unmono32((unsigned)(key >> 32));
    int cls = (int)(idx % K_CLS);
    float off = (float)cls * 20000.0f;
    size_t bo = (size_t)idx * 4, co = (size_t)t * 4;
    float x1 = boxes[bo+0], y1 = boxes[bo+1], x2 = boxes[bo+2], y2 = boxes[bo+3];
    cboxR[co+0]=x1;     cboxR[co+1]=y1;     cboxR[co+2]=x2;     cboxR[co+3]=y2;
    cboxO[co+0]=x1+off; cboxO[co+1]=y1+off; cboxO[co+2]=x2+off; cboxO[co+3]=y2+off;
    cscore[t] = sc; ccls[t] = cls;
    cvalid[t] = (sc > -5e29f) ? 1 : 0;
}

// ---------------- K4: bit-packed suppression matrix via WMMA ---------------
// Wave handles 16 rows; 16x16 C-tile of V_WMMA_F32_16X16X4_F32 provides
// sums[i][j] = area_i + area_j; __ballot packs 2 rows x 16 cols per VGPR.
__global__ __launch_bounds__(256) void k_mask(const float4* __restrict__ cbox4,
                                              unsigned* __restrict__ maskp) {
    __shared__ float4 tile[TJ];
    __shared__ float  tarea[TJ];
    const int lane = threadIdx.x & 31;
    const int wave = threadIdx.x >> 5;
    const int baseI = (blockIdx.x * 8 + wave) * 16;

    float rx1[8], ry1[8], rx2[8], ry2[8], rar[8];
    {
        int half = (lane >= 16) ? 8 : 0;
#pragma unroll
        for (int v = 0; v < 8; ++v) {
            float4 bb = cbox4[baseI + v + half];
            rx1[v]=bb.x; ry1[v]=bb.y; rx2[v]=bb.z; ry2[v]=bb.w;
            rar[v]=fmaxf(bb.z-bb.x,0.f)*fmaxf(bb.w-bb.y,0.f);
        }
    }
    float aArea = 0.f;
    if (lane < 16) {
        float4 bb = cbox4[baseI + lane];
        aArea = fmaxf(bb.z-bb.x,0.f)*fmaxf(bb.w-bb.y,0.f);
    }
    v2f afrag = { aArea, (lane < 16) ? 1.0f : 0.0f };   // A[m][0..1]; K=2,3 -> 0
    v8f czero = {0.f,0.f,0.f,0.f,0.f,0.f,0.f,0.f};

    for (int jt = 0; jt < MAX_CAND; jt += TJ) {
        __syncthreads();
#pragma unroll
        for (int q = 0; q < TJ/256; ++q) {
            int tix = threadIdx.x + q*256;
            copy16_to_lds(&cbox4[jt + tix], &tile[tix]);
        }
        wait_async_copies();
        __syncthreads();
#pragma unroll
        for (int q = 0; q < TJ/256; ++q) {
            int tix = threadIdx.x + q*256;
            float4 bb = tile[tix];
            tarea[tix] = fmaxf(bb.z-bb.x,0.f)*fmaxf(bb.w-bb.y,0.f);
        }
        __syncthreads();

        for (int jg = 0; jg < TJ; jg += 32) {
            unsigned b0[8], b1[8];
#pragma unroll
            for (int s = 0; s < 2; ++s) {
                int col = jg + s*16 + (lane & 15);
                float4 cb = tile[col];
                float  bA = tarea[col];
                v2f bfrag = { (lane < 16) ? 1.0f : 0.0f,
                              (lane < 16) ? bA   : 0.0f };
                v8f sums;
#ifdef HAVE_WMMA4
                sums = __builtin_amdgcn_wmma_f32_16x16x4_f32(
                    false, afrag, false, bfrag, (short)0, czero, false, false);
#else
#pragma unroll
                for (int v = 0; v < 8; ++v) sums[v] = rar[v] + bA;
#endif
#pragma unroll
                for (int v = 0; v < 8; ++v) {
                    float iw = fmaxf(fminf(rx2[v], cb.z) - fmaxf(rx1[v], cb.x), 0.f);
                    float ih = fmaxf(fminf(ry2[v], cb.w) - fmaxf(ry1[v], cb.y), 0.f);
                    float inter = iw * ih;
                    // iou > 0.5  <=>  inter > 0.5*(sum - inter + eps), denom >= eps
                    bool pred = inter > 0.5f * (sums[v] - inter + 1e-7f);
                    unsigned bl = (unsigned)__ballot(pred);
                    if (s == 0) b0[v] = bl; else b1[v] = bl;
                }
            }
            int word = (jt + jg) >> 5;
#pragma unroll
            for (int v = 0; v < 8; ++v) {
                unsigned wlo = (b0[v] & 0xFFFFu) | ((b1[v] & 0xFFFFu) << 16);
                unsigned whi = (b0[v] >> 16)     | (b1[v] & 0xFFFF0000u);
                if (lane == v)     maskp[(size_t)(baseI + v)     * 256 + word] = wlo;
                if (lane == v + 8) maskp[(size_t)(baseI + v + 8) * 256 + word] = whi;
            }
        }
    }
}

// ---------------- K5: greedy NMS over bit matrix ---------------------------
__global__ void k_nms(const unsigned* __restrict__ maskp,
                      const int* __restrict__ cvalid,
                      int* __restrict__ keep) {
    __shared__ unsigned removed[256];
    __shared__ int kflag;
    removed[threadIdx.x] = 0u;
    __syncthreads();
    for (int i = 0; i < MAX_CAND; ++i) {
        if (threadIdx.x == 0) {
            bool sup = (removed[i >> 5] >> (i & 31)) & 1u;
            int kk = (cvalid[i] && !sup) ? 1 : 0;
            keep[i] = kk; kflag = kk;
        }
        __syncthreads();
        if (kflag) removed[threadIdx.x] |= maskp[(size_t)i * 256 + threadIdx.x];
        __syncthreads();
    }
}

// ---------------- K6: top-100 compact + NMS2(0.9) + outputs ----------------
__global__ void k_final(const int* __restrict__ keep,
                        const float* __restrict__ cboxR,
                        const float* __restrict__ cscore,
                        const int* __restrict__ ccls,
                        float* __restrict__ out) {
    __shared__ int sel[TOPK];
    __shared__ float sbx[TOPK][4];
    __shared__ float siou[TOPK * TOPK];
    __shared__ int sup[TOPK];
    __shared__ int sel2[TOPK];
    __shared__ int sm, sk2;
    int t = threadIdx.x;
    if (t == 0) {
        int m = 0;
        for (int i = 0; i < MAX_CAND && m < TOPK; ++i)
            if (keep[i]) sel[m++] = i;
        sm = m;
    }
    if (t < TOPK) sup[t] = 0;
    __syncthreads();
    int m = sm;
    if (t < m) {
        int gi = sel[t];
        sbx[t][0]=cboxR[gi*4+0]; sbx[t][1]=cboxR[gi*4+1];
        sbx[t][2]=cboxR[gi*4+2]; sbx[t][3]=cboxR[gi*4+3];
    }
    __syncthreads();
    for (int p = t; p < TOPK * TOPK; p += blockDim.x) {
        int i = p / TOPK, j = p % TOPK;
        float v = 0.f;
        if (i < m && j < m) {
            float aA = fmaxf(sbx[i][2]-sbx[i][0],0.f)*fmaxf(sbx[i][3]-sbx[i][1],0.f);
            float aB = fmaxf(sbx[j][2]-sbx[j][0],0.f)*fmaxf(sbx[j][3]-sbx[j][1],0.f);
            float iw = fmaxf(fminf(sbx[i][2],sbx[j][2]) - fmaxf(sbx[i][0],sbx[j][0]), 0.f);
            float ih = fmaxf(fminf(sbx[i][3],sbx[j][3]) - fmaxf(sbx[i][1],sbx[j][1]), 0.f);
            float inter = iw * ih;
            v = inter / (aA + aB - inter + 1e-7f);
        }
        siou[p] = v;
    }
    __syncthreads();
    if (t == 0) {
        int k2 = 0;
        for (int i = 0; i < m; ++i) {
            if (!sup[i]) {
                sel2[k2++] = i;
                for (int j = i + 1; j < m; ++j)
                    if (siou[i * TOPK + j] > 0.9f) sup[j] = 1;
            }
        }
        sk2 = k2;
    }
    __syncthreads();
    int k2 = sk2;
    for (int o = t; o < TOPK; o += blockDim.x) {
        if (o < k2) {
            int li = sel2[o];
            int gi = sel[li];
            out[o*4+0]=sbx[li][0]; out[o*4+1]=sbx[li][1];
            out[o*4+2]=sbx[li][2]; out[o*4+3]=sbx[li][3];
            out[400+o] = cscore[gi];
            out[500+o] = (float)ccls[gi];
            out[600+o] = 1.0f;
        } else {
            out[o*4+0]=0.f; out[o*4+1]=0.f; out[o*4+2]=0.f; out[o*4+3]=0.f;
            out[400+o]=0.f; out[500+o]=-1.0f; out[600+o]=0.f;
        }
    }
}

extern "C" void kernel_launch(void* const* d_in, const int* in_sizes, int n_in,
                              void* d_out, int out_size, void* d_ws, size_t ws_size,
                              hipStream_t stream) {
    (void)in_sizes; (void)n_in; (void)out_size; (void)ws_size;
    const float* prop   = (const float*)d_in[0];
    const float* deltas = (const float*)d_in[1];
    const float* logits = (const float*)d_in[2];
    float* out = (float*)d_out;
    char* ws = (char*)d_ws;
    // workspace layout (total ~13.2 MB)
    unsigned long long* keys = (unsigned long long*)(ws + 0);          // 2,097,152 B
    float*    boxes  = (float*)(ws + 2097152);                         // 2,944,000 B
    float*    cboxR  = (float*)(ws + 5041152);                         //   131,072 B (16B aligned)
    float*    cboxO  = (float*)(ws + 5172224);                         //   131,072 B (16B aligned)
    float*    cscore = (float*)(ws + 5303296);                         //    32,768 B
    int*      ccls   = (int*)  (ws + 5336064);                         //    32,768 B
    int*      cvalid = (int*)  (ws + 5368832);                         //    32,768 B
    int*      keep   = (int*)  (ws + 5401600);                         //    32,768 B
    unsigned* maskp  = (unsigned*)(ws + 5434368);                      // 8,388,608 B

    k_decode<<<R_ROWS, 64, 0, stream>>>(prop, deltas, logits, boxes, keys);
    k_pad<<<(NSORT - NFLAT + 255) / 256, 256, 0, stream>>>(keys);
    for (int k = 2; k <= NSORT; k <<= 1)
        for (int j = k >> 1; j > 0; j >>= 1)
            k_bitonic<<<NSORT / 256, 256, 0, stream>>>(keys, j, k);
    k_gather<<<MAX_CAND / 256, 256, 0, stream>>>(keys, boxes, cboxR, cboxO,
                                                 cscore, ccls, cvalid);
    k_mask<<<MAX_CAND / 128, 256, 0, stream>>>((const float4*)cboxO, maskp);
    k_nms<<<1, 256, 0, stream>>>(maskp, cvalid, keep);
    k_final<<<1, 128, 0, stream>>>(keep, cboxR, cscore, ccls, out);
}